// Model_42941083025657
// MI455X (gfx1250) — compile-verified
//
#include <hip/hip_runtime.h>
#include <hip/hip_bf16.h>
#include <cstdint>

#define SEQ   2048
#define DIM   300      // embedding dim == H1 == 2*UNITS
#define UNITS 150
#define GATES 600      // 4*UNITS
#define NSYN  4

typedef __attribute__((ext_vector_type(2))) float     v2f;
typedef __attribute__((ext_vector_type(8))) float     v8f;
typedef __attribute__((ext_vector_type(2))) _Float16  h2v;

// ---------------- embedding gather ----------------
__global__ void __launch_bounds__(320) k_gather(const int* __restrict__ sent,
                                                const float* __restrict__ E,
                                                float* __restrict__ emb) {
  const int s = blockIdx.x, d = threadIdx.x;
  if (d < DIM) emb[(size_t)s * DIM + d] = E[(size_t)sent[s] * DIM + d];
}

// ---------------- f32 WMMA GEMM: C[M,N] = A[M,K] @ B[K,N] + bias ----------------
// One wave computes one 16x16 tile with V_WMMA_F32_16X16X4_F32 (K stepped by 4).
// A layout : lane m = lane&15, VGPR pair holds K = 2*(lane>>4) + {0,1} (+kb)
// B layout : lane n = lane&15, VGPR pair holds K = 2*(lane>>4) + {0,1} (+kb)
// C layout : VGPR r -> row m0 + 8*(lane>>4) + r, col n0 + (lane&15)
__global__ void __launch_bounds__(32) k_wmma_gemm_f32(
    const float* __restrict__ A, int lda,
    const float* __restrict__ B, int ldb,
    const float* __restrict__ bias,
    float* __restrict__ C, int ldc,
    int N, int K)
{
  const int lane = threadIdx.x;
  const int lo = lane & 15, hi = lane >> 4;
  const int m = blockIdx.x * 16 + lo;
  const int n = blockIdx.y * 16 + lo;
  const int nl = (n < N) ? n : (N - 1);          // clamp loads on partial N tile
  const float* Ap = A + (size_t)m * lda + 2 * hi;
  const float* Bp = B + (size_t)(2 * hi) * ldb + nl;
  v8f acc = {};
  for (int kb = 0; kb < K; kb += 4) {
    v2f a, b;
    a.x = Ap[kb];
    a.y = Ap[kb + 1];
    b.x = Bp[(size_t)kb * ldb];
    b.y = Bp[(size_t)kb * ldb + ldb];
    acc = __builtin_amdgcn_wmma_f32_16x16x4_f32(false, a, false, b,
                                                (short)0, acc, false, false);
  }
  if (n < N) {
    const float bv = bias[n];
#pragma unroll
    for (int r = 0; r < 8; ++r) {
      const int mr = blockIdx.x * 16 + hi * 8 + r;
      C[(size_t)mr * ldc + n] = acc[r] + bv;
    }
  }
}

// ---------------- serial LSTM scan (1 WGP per direction) ----------------
__device__ __forceinline__ float sigmoidf_(float x) {
  return 1.0f / (1.0f + __expf(-x));
}

__device__ __forceinline__ float dot8h(uint4 w, uint4 h, float acc) {
  union { uint4 u; h2v p[4]; } uw, uh;
  uw.u = w; uh.u = h;
#pragma unroll
  for (int i = 0; i < 4; ++i) {
    acc = fmaf((float)uw.p[i].x, (float)uh.p[i].x, acc);
    acc = fmaf((float)uw.p[i].y, (float)uh.p[i].y, acc);
  }
  return acc;
}

#define COLH 152   // halfs per Wr column: 150 + 2 zero pad (19 x b128 per column)
#define SCAN_LDS_BYTES (GATES * COLH * 2 + COLH * 2 + GATES * 4)

__global__ void __launch_bounds__(640) k_lstm_scan(
    const float* __restrict__ Xf, const float* __restrict__ Xb,
    const float* __restrict__ Wrf, const float* __restrict__ Wrb,
    float* __restrict__ hidden)
{
  extern __shared__ char smem[];
  _Float16* Wc  = (_Float16*)smem;                          // [600][152] col-major f16
  _Float16* hsh = (_Float16*)(smem + GATES * COLH * 2);     // [152] h broadcast (f16)
  float*    z   = (float*)(smem + GATES * COLH * 2 + COLH * 2); // [600]

  const int dir = blockIdx.x;
  const float* X  = dir ? Xb  : Xf;
  const float* Wr = dir ? Wrb : Wrf;
  const int tid = threadIdx.x;

  // stage recurrent weights into LDS (f16, column-major, zero-padded K)
  for (int e = tid; e < UNITS * GATES; e += 640) {
    const int k = e / GATES, j = e - k * GATES;
    Wc[j * COLH + k] = (_Float16)Wr[e];
  }
  for (int j = tid; j < GATES; j += 640) {
    Wc[j * COLH + 150] = (_Float16)0.f;
    Wc[j * COLH + 151] = (_Float16)0.f;
  }
  if (tid < COLH) hsh[tid] = (_Float16)0.f;
  __syncthreads();

  float c = 0.0f;   // per-unit cell state lives in a register of thread tid<150
  for (int t = 0; t < SEQ; ++t) {
    const int row = dir ? (SEQ - 1 - t) : t;
    if (tid < GATES) {
      float acc = X[(size_t)row * GATES + tid];            // precomputed emb@Wk + b
      const uint4* wp = (const uint4*)(Wc + tid * COLH);
      const uint4* hp = (const uint4*)hsh;
#pragma unroll
      for (int q = 0; q < COLH / 8; ++q)
        acc = dot8h(wp[q], hp[q], acc);
      z[tid] = acc;
    }
    __syncthreads();
    if (tid < UNITS) {
      const float ig = sigmoidf_(z[tid]);
      const float fg = sigmoidf_(z[UNITS + tid]);
      const float gg = tanhf(z[2 * UNITS + tid]);
      const float og = sigmoidf_(z[3 * UNITS + tid]);
      c = fg * c + ig * gg;
      const float h = og * tanhf(c);
      hsh[tid] = (_Float16)h;
      hidden[(size_t)row * (2 * UNITS) + dir * UNITS + tid] = h;
    }
    __syncthreads();
  }
}

// ---------------- synonym attention (one block per position) ----------------
__global__ void __launch_bounds__(256) k_attn(
    const int* __restrict__ sent, const int* __restrict__ syn,
    const float* __restrict__ E, const float* __restrict__ outw,
    const float* __restrict__ hidden,
    const float* __restrict__ W2, const float* __restrict__ b2,
    float* __restrict__ hhat, float* __restrict__ c2)
{
  __shared__ float red[256];
  __shared__ float coeff[NSYN];
  __shared__ int   rows[NSYN];
  const int s = blockIdx.x, tid = threadIdx.x;
  const int prev = (s == 0) ? 0 : (s - 1);
  if (tid < NSYN) rows[tid] = syn[(size_t)sent[s] * NSYN + tid];
  __syncthreads();

  // coeff[k] = exp( dot(syn_emb[s,k,:], out[prev,:]) ), 64 threads per k
  {
    const int k = tid >> 6, l = tid & 63;
    const float* er = E + (size_t)rows[k] * DIM;
    const float* op = outw + (size_t)prev * DIM;
    float p = 0.f;
    for (int d = l; d < DIM; d += 64) p += er[d] * op[d];
    red[tid] = p;
    __syncthreads();
    for (int off = 32; off > 0; off >>= 1) {
      if (l < off) red[tid] += red[tid + off];
      __syncthreads();
    }
    if (l == 0) coeff[k] = __expf(red[tid]);
  }
  __syncthreads();

  // h_hat = sum_k coeff_k * syn_emb + hidden[prev];  c2 = exp(tanh(h_hat@W2 + b2))
  float acc = 0.f;
  for (int d = tid; d < DIM; d += 256) {
    const float m = coeff[0] * E[(size_t)rows[0] * DIM + d]
                  + coeff[1] * E[(size_t)rows[1] * DIM + d]
                  + coeff[2] * E[(size_t)rows[2] * DIM + d]
                  + coeff[3] * E[(size_t)rows[3] * DIM + d];
    const float hh = m + hidden[(size_t)prev * DIM + d];
    hhat[(size_t)s * DIM + d] = hh;
    acc += hh * W2[d];
  }
  red[tid] = acc;
  __syncthreads();
  for (int off = 128; off > 0; off >>= 1) {
    if (tid < off) red[tid] += red[tid + off];
    __syncthreads();
  }
  if (tid == 0) c2[s] = __expf(tanhf(red[0] + b2[0]));
}

// ---------------- final weighted reduction + heads ----------------
__global__ void __launch_bounds__(512) k_final(
    const float* __restrict__ c2, const float* __restrict__ hhat,
    const float* __restrict__ We, const float* __restrict__ be,
    const float* __restrict__ Ws, const float* __restrict__ bs,
    float* __restrict__ out9)
{
  __shared__ float Hs[DIM];
  const int tid = threadIdx.x;
  if (tid < DIM) {
    float a = 0.f;
    for (int s = 0; s < SEQ; ++s)
      a = fmaf(c2[s], hhat[(size_t)s * DIM + tid], a);
    Hs[tid] = a;
  }
  __syncthreads();
  if (tid < 8) {
    float a = be[tid];
    for (int d = 0; d < DIM; ++d) a = fmaf(Hs[d], We[d * 8 + tid], a);
    out9[tid] = a;                 // emotion logits [1,8]
  } else if (tid == 8) {
    float a = bs[0];
    for (int d = 0; d < DIM; ++d) a = fmaf(Hs[d], Ws[d], a);
    out9[8] = a;                   // sentiment logit [1,1]
  }
}

extern "C" void kernel_launch(void* const* d_in, const int* in_sizes, int n_in,
                              void* d_out, int out_size, void* d_ws, size_t ws_size,
                              hipStream_t stream) {
  (void)in_sizes; (void)n_in; (void)out_size; (void)ws_size;
  const int*   sent = (const int*)d_in[0];
  const int*   syn  = (const int*)d_in[1];
  const float* E    = (const float*)d_in[2];
  const float* Wk_f = (const float*)d_in[3];
  const float* Wr_f = (const float*)d_in[4];
  const float* b_f  = (const float*)d_in[5];
  const float* Wk_b = (const float*)d_in[6];
  const float* Wr_b = (const float*)d_in[7];
  const float* b_b  = (const float*)d_in[8];
  const float* W1   = (const float*)d_in[9];
  const float* b1   = (const float*)d_in[10];
  const float* W2   = (const float*)d_in[11];
  const float* b2   = (const float*)d_in[12];
  const float* We   = (const float*)d_in[13];
  const float* be   = (const float*)d_in[14];
  const float* Ws   = (const float*)d_in[15];
  const float* bs   = (const float*)d_in[16];
  float* out9 = (float*)d_out;

  char* base = (char*)d_ws;
  size_t off = 0;
  auto alloc = [&](size_t bytes) -> void* {
    off = (off + 255) & ~(size_t)255;
    void* p = base + off;
    off += bytes;
    return p;
  };
  float* emb    = (float*)alloc((size_t)SEQ * DIM * 4);    // gathered embeddings
  float* Xf     = (float*)alloc((size_t)SEQ * GATES * 4);  // emb@Wk_f + b_f
  float* Xb     = (float*)alloc((size_t)SEQ * GATES * 4);  // emb@Wk_b + b_b
  float* hidden = (float*)alloc((size_t)SEQ * DIM * 4);    // [hf|hb]
  float* outw   = (float*)alloc((size_t)SEQ * DIM * 4);    // hidden@W1 + b1
  float* hhat   = (float*)alloc((size_t)SEQ * DIM * 4);
  float* c2b    = (float*)alloc((size_t)SEQ * 4);

  k_gather<<<SEQ, 320, 0, stream>>>(sent, E, emb);

  k_wmma_gemm_f32<<<dim3(SEQ / 16, (GATES + 15) / 16), 32, 0, stream>>>(
      emb, DIM, Wk_f, GATES, b_f, Xf, GATES, GATES, DIM);
  k_wmma_gemm_f32<<<dim3(SEQ / 16, (GATES + 15) / 16), 32, 0, stream>>>(
      emb, DIM, Wk_b, GATES, b_b, Xb, GATES, GATES, DIM);

  (void)hipFuncSetAttribute((const void*)k_lstm_scan,
                            hipFuncAttributeMaxDynamicSharedMemorySize,
                            SCAN_LDS_BYTES);
  k_lstm_scan<<<2, 640, SCAN_LDS_BYTES, stream>>>(Xf, Xb, Wr_f, Wr_b, hidden);

  k_wmma_gemm_f32<<<dim3(SEQ / 16, (DIM + 15) / 16), 32, 0, stream>>>(
      hidden, DIM, W1, DIM, b1, outw, DIM, DIM, DIM);

  k_attn<<<SEQ, 256, 0, stream>>>(sent, syn, E, outw, hidden, W2, b2, hhat, c2b);
  k_final<<<1, 512, 0, stream>>>(c2b, hhat, We, be, Ws, bs, out9);
}